// GATE_81346680586566
// MI455X (gfx1250) — compile-verified
//
#include <hip/hip_runtime.h>
#include <math.h>

// Problem constants (match the JAX reference)
#define N_NODES 10000
#define N_EDGES 320000
#define F_INDIM 512
#define HIDDIM  256
#define OUTDIM  32
#define NEG_SLOPE 0.2f

typedef __attribute__((ext_vector_type(2))) float v2f;
typedef __attribute__((ext_vector_type(8))) float v8f;

__device__ __forceinline__ float leaky(float v) {
    return v > 0.f ? v : NEG_SLOPE * v;
}

// float atomic max via signed-max / unsigned-min bit trick (init must be -inf)
__device__ __forceinline__ void atomic_max_f32(float* addr, float v) {
    if (v >= 0.f) atomicMax((int*)addr, __float_as_int(v));
    else          atomicMin((unsigned int*)addr, __float_as_uint(v));
}

// -------------------- fill --------------------
__global__ void fill_kernel(float* __restrict__ p, float val, int n) {
    int i = blockIdx.x * blockDim.x + threadIdx.x;
    if (i < n) p[i] = val;
}

// -------------------- generic f32 WMMA GEMM: D[M,N] = A[M,K] @ B[K,N] --------------------
// One wave32 per 16x16 output tile; K stepped by 4 via V_WMMA_F32_16X16X4_F32.
// A-layout per ISA: lanes 0-15 -> M=lane, K={k0,k0+1}; lanes 16-31 -> M=lane-16, K={k0+2,k0+3}.
__global__ void wmma_gemm_f32(const float* __restrict__ A, const float* __restrict__ B,
                              float* __restrict__ D, int M, int N, int K, int numTiles) {
    int wave = threadIdx.x >> 5;
    int tile = blockIdx.x * (blockDim.x >> 5) + wave;
    if (tile >= numTiles) return;
    int ntN  = N >> 4;
    int tm   = (tile / ntN) << 4;
    int tn   = (tile % ntN) << 4;
    int lane = threadIdx.x & 31;
    int half = lane >> 4;          // 0: lanes 0-15, 1: lanes 16-31
    int l15  = lane & 15;

    v8f c = {0.f,0.f,0.f,0.f,0.f,0.f,0.f,0.f};
    const float* arow = A + (size_t)(tm + l15) * K;
    for (int k0 = 0; k0 < K; k0 += 4) {
        int kk = k0 + 2 * half;
        v2f a = *(const v2f*)(arow + kk);                 // A[tm+l15][kk], A[tm+l15][kk+1]
        v2f b;
        b.x = B[(size_t)kk       * N + tn + l15];         // B[kk  ][tn+l15]
        b.y = B[(size_t)(kk + 1) * N + tn + l15];         // B[kk+1][tn+l15]
        c = __builtin_amdgcn_wmma_f32_16x16x4_f32(false, a, false, b, (short)0, c, false, false);
    }
    // C/D layout: VGPR v -> M = tm + v + 8*half, N = tn + l15
    float* drow = D + (size_t)(tm + half * 8) * N + tn + l15;
#pragma unroll
    for (int v = 0; v < 8; ++v) drow[(size_t)v * N] = c[v];
}

// -------------------- decoder: Out = sigmoid(Z @ Z^T), Z:[Nn, Kd] --------------------
// Z (1.28 MB) is re-read ~625x per tile-row: keep it resident in the 192 MB L2 (regular
// RT loads). The 400 MB output is write-once/never-read: stream it past the caches with
// non-temporal stores so it does not evict Z from L2.
__global__ void wmma_zzt_sigmoid(const float* __restrict__ Z, float* __restrict__ Out,
                                 int Nn, int Kd, int numTiles) {
    int wave = threadIdx.x >> 5;
    int tile = blockIdx.x * (blockDim.x >> 5) + wave;
    if (tile >= numTiles) return;
    int ntN  = Nn >> 4;
    int tm   = (tile / ntN) << 4;
    int tn   = (tile % ntN) << 4;
    int lane = threadIdx.x & 31;
    int half = lane >> 4;
    int l15  = lane & 15;

    v8f c = {0.f,0.f,0.f,0.f,0.f,0.f,0.f,0.f};
    const float* arow = Z + (size_t)(tm + l15) * Kd;
    const float* brow = Z + (size_t)(tn + l15) * Kd;      // B[k][n] = Z[n][k] -> row of Z
#pragma unroll
    for (int k0 = 0; k0 < OUTDIM; k0 += 4) {
        int kk = k0 + 2 * half;
        v2f a = *(const v2f*)(arow + kk);
        v2f b = *(const v2f*)(brow + kk);
        c = __builtin_amdgcn_wmma_f32_16x16x4_f32(false, a, false, b, (short)0, c, false, false);
    }
    float* drow = Out + (size_t)(tm + half * 8) * Nn + tn + l15;
#pragma unroll
    for (int v = 0; v < 8; ++v) {
        float s = 1.f / (1.f + __expf(-c[v]));
        __builtin_nontemporal_store(s, drow + (size_t)v * Nn);
    }
}

// -------------------- per-node attention scores: ss[n]=h[n].a_src, sd[n]=h[n].a_dst --------------------
__global__ void scores_kernel(const float* __restrict__ h, const float* __restrict__ a_src,
                              const float* __restrict__ a_dst, float* __restrict__ ss,
                              float* __restrict__ sd, int Nn, int Dim) {
    int node = blockIdx.x * (blockDim.x >> 5) + (threadIdx.x >> 5);
    if (node >= Nn) return;
    int lane = threadIdx.x & 31;
    float s1 = 0.f, s2 = 0.f;
    for (int d = lane; d < Dim; d += 32) {
        float hv = h[(size_t)node * Dim + d];
        s1 += hv * a_src[d];
        s2 += hv * a_dst[d];
    }
#pragma unroll
    for (int off = 16; off > 0; off >>= 1) {
        s1 += __shfl_xor(s1, off, 32);
        s2 += __shfl_xor(s2, off, 32);
    }
    if (lane == 0) { ss[node] = s1; sd[node] = s2; }
}

// -------------------- segment max over dst --------------------
__global__ void edge_max_kernel(const int* __restrict__ src, const int* __restrict__ dst,
                                const float* __restrict__ ss, const float* __restrict__ sd,
                                float* __restrict__ m, int E) {
    int e = blockIdx.x * blockDim.x + threadIdx.x;
    if (e >= E) return;
    float v = leaky(ss[src[e]] + sd[dst[e]]);
    atomic_max_f32(&m[dst[e]], v);
}

// -------------------- ex[e] = exp(e - m[dst]); denom[dst] += ex[e] --------------------
__global__ void edge_exp_kernel(const int* __restrict__ src, const int* __restrict__ dst,
                                const float* __restrict__ ss, const float* __restrict__ sd,
                                const float* __restrict__ m, float* __restrict__ denom,
                                float* __restrict__ ex, int E) {
    int e = blockIdx.x * blockDim.x + threadIdx.x;
    if (e >= E) return;
    int d = dst[e];
    float v = leaky(ss[src[e]] + sd[d]);
    float xv = __expf(v - m[d]);
    ex[e] = xv;
    atomicAdd(&denom[d], xv);
}

// -------------------- out[dst] += h[src] * (ex[e] / (denom[dst]+1e-16)) --------------------
// one thread handles 4 features of one edge
__global__ void aggregate_kernel(const int* __restrict__ src, const int* __restrict__ dst,
                                 const float* __restrict__ h, const float* __restrict__ ex,
                                 const float* __restrict__ denom, float* __restrict__ out,
                                 int E, int Dim) {
    int chunks = Dim >> 2;
    int idx = blockIdx.x * blockDim.x + threadIdx.x;
    if (idx >= E * chunks) return;
    int e  = idx / chunks;
    int f4 = (idx % chunks) << 2;
    int s = src[e], d = dst[e];
    float alpha = __builtin_nontemporal_load(ex + e) / (denom[d] + 1e-16f);
    const float4 hv = *(const float4*)(h + (size_t)s * Dim + f4);
    float* o = out + (size_t)d * Dim + f4;
    atomicAdd(o + 0, hv.x * alpha);
    atomicAdd(o + 1, hv.y * alpha);
    atomicAdd(o + 2, hv.z * alpha);
    atomicAdd(o + 3, hv.w * alpha);
}

// -------------------- out = (relu?) (out + b) --------------------
__global__ void bias_act_kernel(float* __restrict__ p, const float* __restrict__ b,
                                int total, int Dim, int do_relu) {
    int i = blockIdx.x * blockDim.x + threadIdx.x;
    if (i >= total) return;
    float v = p[i] + b[i % Dim];
    if (do_relu) v = fmaxf(v, 0.f);
    p[i] = v;
}

extern "C" void kernel_launch(void* const* d_in, const int* in_sizes, int n_in,
                              void* d_out, int out_size, void* d_ws, size_t ws_size,
                              hipStream_t stream) {
    const float* x   = (const float*)d_in[0];
    const int*   ei  = (const int*)  d_in[1];
    const float* W1  = (const float*)d_in[2];
    const float* a1s = (const float*)d_in[3];
    const float* a1d = (const float*)d_in[4];
    const float* b1  = (const float*)d_in[5];
    const float* W2  = (const float*)d_in[6];
    const float* a2s = (const float*)d_in[7];
    const float* a2d = (const float*)d_in[8];
    const float* b2  = (const float*)d_in[9];
    float* out = (float*)d_out;

    const int* src = ei;            // edge_index[0]
    const int* dst = ei + N_EDGES;  // edge_index[1]

    // workspace layout (floats): ~24.8 MB total
    float* ws  = (float*)d_ws;
    float* h1  = ws;                                  // [N, HID]
    float* g1  = h1  + (size_t)N_NODES * HIDDIM;      // [N, HID]
    float* h2  = g1  + (size_t)N_NODES * HIDDIM;      // [N, OUT]
    float* Z   = h2  + (size_t)N_NODES * OUTDIM;      // [N, OUT]
    float* ex  = Z   + (size_t)N_NODES * OUTDIM;      // [E]
    float* ss1 = ex  + N_EDGES;
    float* sd1 = ss1 + N_NODES;
    float* m1  = sd1 + N_NODES;
    float* dn1 = m1  + N_NODES;
    float* ss2 = dn1 + N_NODES;
    float* sd2 = ss2 + N_NODES;
    float* m2  = sd2 + N_NODES;
    float* dn2 = m2  + N_NODES;

    const float NEG_INF = -__builtin_huge_valf();
    const int TB = 256;

    // ---------------- layer 1: h1 = x @ W1 ----------------
    {
        int tiles = (N_NODES / 16) * (HIDDIM / 16);   // 625*16 = 10000
        wmma_gemm_f32<<<(tiles + 3) / 4, 128, 0, stream>>>(x, W1, h1, N_NODES, HIDDIM, F_INDIM, tiles);
    }
    scores_kernel<<<(N_NODES + 3) / 4, 128, 0, stream>>>(h1, a1s, a1d, ss1, sd1, N_NODES, HIDDIM);
    fill_kernel<<<(N_NODES + TB - 1) / TB, TB, 0, stream>>>(m1, NEG_INF, N_NODES);
    fill_kernel<<<(N_NODES + TB - 1) / TB, TB, 0, stream>>>(dn1, 0.f, N_NODES);
    fill_kernel<<<(N_NODES * HIDDIM + TB - 1) / TB, TB, 0, stream>>>(g1, 0.f, N_NODES * HIDDIM);
    edge_max_kernel<<<(N_EDGES + TB - 1) / TB, TB, 0, stream>>>(src, dst, ss1, sd1, m1, N_EDGES);
    edge_exp_kernel<<<(N_EDGES + TB - 1) / TB, TB, 0, stream>>>(src, dst, ss1, sd1, m1, dn1, ex, N_EDGES);
    {
        int work = N_EDGES * (HIDDIM / 4);            // 20.48M
        aggregate_kernel<<<(work + TB - 1) / TB, TB, 0, stream>>>(src, dst, h1, ex, dn1, g1, N_EDGES, HIDDIM);
    }
    bias_act_kernel<<<(N_NODES * HIDDIM + TB - 1) / TB, TB, 0, stream>>>(g1, b1, N_NODES * HIDDIM, HIDDIM, 1);

    // ---------------- layer 2: h2 = g1 @ W2 ----------------
    {
        int tiles = (N_NODES / 16) * (OUTDIM / 16);   // 625*2 = 1250
        wmma_gemm_f32<<<(tiles + 3) / 4, 128, 0, stream>>>(g1, W2, h2, N_NODES, OUTDIM, HIDDIM, tiles);
    }
    scores_kernel<<<(N_NODES + 3) / 4, 128, 0, stream>>>(h2, a2s, a2d, ss2, sd2, N_NODES, OUTDIM);
    fill_kernel<<<(N_NODES + TB - 1) / TB, TB, 0, stream>>>(m2, NEG_INF, N_NODES);
    fill_kernel<<<(N_NODES + TB - 1) / TB, TB, 0, stream>>>(dn2, 0.f, N_NODES);
    fill_kernel<<<(N_NODES * OUTDIM + TB - 1) / TB, TB, 0, stream>>>(Z, 0.f, N_NODES * OUTDIM);
    edge_max_kernel<<<(N_EDGES + TB - 1) / TB, TB, 0, stream>>>(src, dst, ss2, sd2, m2, N_EDGES);
    edge_exp_kernel<<<(N_EDGES + TB - 1) / TB, TB, 0, stream>>>(src, dst, ss2, sd2, m2, dn2, ex, N_EDGES);
    {
        int work = N_EDGES * (OUTDIM / 4);            // 2.56M
        aggregate_kernel<<<(work + TB - 1) / TB, TB, 0, stream>>>(src, dst, h2, ex, dn2, Z, N_EDGES, OUTDIM);
    }
    bias_act_kernel<<<(N_NODES * OUTDIM + TB - 1) / TB, TB, 0, stream>>>(Z, b2, N_NODES * OUTDIM, OUTDIM, 0);

    // ---------------- decoder: out = sigmoid(Z @ Z^T) ----------------
    {
        int tiles = (N_NODES / 16) * (N_NODES / 16);  // 390625
        wmma_zzt_sigmoid<<<(tiles + 3) / 4, 128, 0, stream>>>(Z, out, N_NODES, OUTDIM, tiles);
    }
}